// GCN_17532056502398
// MI455X (gfx1250) — compile-verified
//
#include <hip/hip_runtime.h>
#include <hip/hip_bf16.h>

#define N_NODES  100000
#define N_EDGES  300000
#define N_GRAPHS 4096
#define EMB      256
#define NLAYERS  5
#define EPSV     1e-5f

typedef float v2f __attribute__((ext_vector_type(2)));
typedef float v8f __attribute__((ext_vector_type(8)));

// ---------------------------------------------------------------- utilities
__global__ void gcn_zero4(float4* __restrict__ p, int n4) {
    const float4 z = make_float4(0.f, 0.f, 0.f, 0.f);
    for (int i = blockIdx.x * blockDim.x + threadIdx.x; i < n4;
         i += gridDim.x * blockDim.x)
        p[i] = z;
}

// ------------------------------------------------------------ embeddings
// h[m,c] = sum_k x[m,k]*W[c,k] + b[c]   (K=40), one block per node
__global__ void __launch_bounds__(EMB)
gcn_node_embed(const float* __restrict__ x, const float* __restrict__ W,
               const float* __restrict__ b, float* __restrict__ h) {
    __shared__ float sx[40];
    int m = blockIdx.x, c = threadIdx.x;
    if (c < 40) sx[c] = x[(long)m * 40 + c];
    __syncthreads();
    float acc = b[c];
    #pragma unroll
    for (int k = 0; k < 40; ++k) acc += sx[k] * W[c * 40 + k];
    h[(long)m * EMB + c] = acc;
}

// ea[e,c] = sum_k ea_in[e,k]*W[c,k] + b[c]   (K=10), one block per edge
__global__ void __launch_bounds__(EMB)
gcn_edge_embed(const float* __restrict__ ein, const float* __restrict__ W,
               const float* __restrict__ b, float* __restrict__ ea) {
    __shared__ float se[10];
    int e = blockIdx.x, c = threadIdx.x;
    if (c < 10) se[c] = ein[(long)e * 10 + c];
    __syncthreads();
    float acc = b[c];
    #pragma unroll
    for (int k = 0; k < 10; ++k) acc += se[k] * W[c * 10 + k];
    ea[(long)e * EMB + c] = acc;
}

// ------------------------------------------------------------ degrees
__global__ void gcn_degree(const int* __restrict__ ei, float* __restrict__ deg) {
    int e = blockIdx.x * blockDim.x + threadIdx.x;
    if (e < N_EDGES) atomicAdd(&deg[ei[e]], 1.0f);   // row = ei[0][e]
}

__global__ void gcn_dinv(float* __restrict__ deg, float* __restrict__ dinv) {
    int i = blockIdx.x * blockDim.x + threadIdx.x;
    if (i < N_NODES) {
        float d = deg[i] + 1.0f;
        deg[i]  = d;              // deg buffer now holds deg+1
        dinv[i] = rsqrtf(d);
    }
}

// ------------------------------------------------------------ FP32 WMMA GEMM
// hx = h @ W^T + bias.  hx[m,n] = sum_k h[m,k]*W[n,k].
// Block: 256 threads (8 waves), owns rows m0..m0+31, all 256 columns.
// Each wave holds 4 accumulators (2 M-tiles x 2 N-tiles): every B fetch and
// every A LDS read feeds two V_WMMA_F32_16X16X4_F32 ops.
__global__ void __launch_bounds__(256)
gcn_gemm_wmma(const float* __restrict__ h, const float* __restrict__ W,
              const float* __restrict__ bias, float* __restrict__ hx) {
    __shared__ float sA[32][260];          // +4 pad: rows hit distinct banks
    const int tid = threadIdx.x;
    const int m0  = blockIdx.x * 32;

    // stage 32x256 A tile: b128 global loads -> b128 LDS stores
    {
        const int colb = (tid & 63) * 4;       // 16B-aligned within row
        const int rsub = tid >> 6;             // 0..3
        #pragma unroll
        for (int r = 0; r < 8; ++r) {
            const int row = r * 4 + rsub;
            const float4 v = *(const float4*)(h + (long)(m0 + row) * EMB + colb);
            *(float4*)&sA[row][colb] = v;
        }
    }
    __syncthreads();

    const int wave  = tid >> 5;
    const int lane  = tid & 31;
    const int mrow  = lane & 15;           // M (for A) / N (for B) within tile
    const int khalf = (lane >> 4) << 1;    // lanes 0-15 -> K+0,+1 ; 16-31 -> K+2,+3
    const int n0    = wave * 32;           // this wave's first N tile

    const float* __restrict__ w0 = W + (long)(n0 + mrow) * EMB;
    const float* __restrict__ w1 = W + (long)(n0 + 16 + mrow) * EMB;

    v8f acc00 = {}; v8f acc01 = {};        // M-tile 0 x {N0, N1}
    v8f acc10 = {}; v8f acc11 = {};        // M-tile 1 x {N0, N1}
    #pragma unroll 4
    for (int k0 = 0; k0 < EMB; k0 += 4) {
        const int kb = k0 + khalf;
        v2f a0, a1, b0, b1;
        a0.x = sA[mrow][kb];        a0.y = sA[mrow][kb + 1];
        a1.x = sA[16 + mrow][kb];   a1.y = sA[16 + mrow][kb + 1];
        b0.x = w0[kb];              b0.y = w0[kb + 1];
        b1.x = w1[kb];              b1.y = w1[kb + 1];
        acc00 = __builtin_amdgcn_wmma_f32_16x16x4_f32(
                    false, a0, false, b0, (short)0, acc00, false, false);
        acc01 = __builtin_amdgcn_wmma_f32_16x16x4_f32(
                    false, a0, false, b1, (short)0, acc01, false, false);
        acc10 = __builtin_amdgcn_wmma_f32_16x16x4_f32(
                    false, a1, false, b0, (short)0, acc10, false, false);
        acc11 = __builtin_amdgcn_wmma_f32_16x16x4_f32(
                    false, a1, false, b1, (short)0, acc11, false, false);
    }

    // D layout: VGPR r -> M = r (lanes 0-15) / r+8 (lanes 16-31); N = lane&15
    const int colA  = n0 + mrow;
    const int colB  = n0 + 16 + mrow;
    const int rbase = (lane >> 4) * 8;
    const float bA = bias[colA], bB = bias[colB];
    #pragma unroll
    for (int r = 0; r < 8; ++r) {
        const long ma = m0 + rbase + r;
        const long mb = m0 + 16 + rbase + r;
        hx[ma * EMB + colA] = acc00[r] + bA;
        hx[ma * EMB + colB] = acc01[r] + bB;
        hx[mb * EMB + colA] = acc10[r] + bA;
        hx[mb * EMB + colB] = acc11[r] + bB;
    }
}

// ------------------------------------------------------------ edge messages
// agg[col] += dinv[row]*dinv[col] * relu(hx[row] + ea[e])
// 4 edges per 256-thread block; 64 lanes x float4 channels per edge (b128
// gathers -> widest possible use of the 23.3 TB/s HBM path).
__global__ void __launch_bounds__(256)
gcn_edge_msg(const int* __restrict__ ei, const float* __restrict__ dinv,
             const float* __restrict__ hx, const float* __restrict__ ea,
             float* __restrict__ agg) {
    const int e  = blockIdx.x * 4 + (threadIdx.x >> 6);
    const int cb = (threadIdx.x & 63) * 4;
    const int r   = ei[e];
    const int col = ei[N_EDGES + e];
    const float nrm = dinv[r] * dinv[col];
    const float4 hv = *(const float4*)(hx + (long)r * EMB + cb);
    const float4 ev = *(const float4*)(ea + (long)e * EMB + cb);
    float* dst = agg + (long)col * EMB + cb;
    atomicAdd(dst + 0, nrm * fmaxf(hv.x + ev.x, 0.f));
    atomicAdd(dst + 1, nrm * fmaxf(hv.y + ev.y, 0.f));
    atomicAdd(dst + 2, nrm * fmaxf(hv.z + ev.z, 0.f));
    atomicAdd(dst + 3, nrm * fmaxf(hv.w + ev.w, 0.f));
}

// ------------------------------------------------------------ combine + BN stats
// t = h + agg + relu(hx + root)/deg ; write t into agg in place; accumulate
// per-channel sum / sumsq.  64 nodes per block, 64 threads x float4 channels.
__global__ void __launch_bounds__(64)
gcn_combine_stats(const float* __restrict__ h, const float* __restrict__ hx,
                  float* __restrict__ agg, const float* __restrict__ root,
                  const float* __restrict__ deg, float* __restrict__ stats) {
    const int cb = threadIdx.x * 4;        // channels cb..cb+3
    const int m0 = blockIdx.x * 64;
    const float4 rt = *(const float4*)(root + cb);
    float4 s  = make_float4(0.f, 0.f, 0.f, 0.f);
    float4 ss = make_float4(0.f, 0.f, 0.f, 0.f);
    for (int i = 0; i < 64; ++i) {
        const int m = m0 + i;
        if (m >= N_NODES) break;
        const long off = (long)m * EMB + cb;
        const float rd = 1.0f / deg[m];
        const float4 hv = *(const float4*)(h + off);
        const float4 av = *(const float4*)(agg + off);
        const float4 xv = *(const float4*)(hx + off);
        float4 t;
        t.x = hv.x + av.x + fmaxf(xv.x + rt.x, 0.f) * rd;
        t.y = hv.y + av.y + fmaxf(xv.y + rt.y, 0.f) * rd;
        t.z = hv.z + av.z + fmaxf(xv.z + rt.z, 0.f) * rd;
        t.w = hv.w + av.w + fmaxf(xv.w + rt.w, 0.f) * rd;
        *(float4*)(agg + off) = t;
        s.x += t.x; s.y += t.y; s.z += t.z; s.w += t.w;
        ss.x += t.x * t.x; ss.y += t.y * t.y;
        ss.z += t.z * t.z; ss.w += t.w * t.w;
    }
    atomicAdd(&stats[cb + 0], s.x);
    atomicAdd(&stats[cb + 1], s.y);
    atomicAdd(&stats[cb + 2], s.z);
    atomicAdd(&stats[cb + 3], s.w);
    atomicAdd(&stats[EMB + cb + 0], ss.x);
    atomicAdd(&stats[EMB + cb + 1], ss.y);
    atomicAdd(&stats[EMB + cb + 2], ss.z);
    atomicAdd(&stats[EMB + cb + 3], ss.w);
}

__global__ void __launch_bounds__(EMB)
gcn_bn_finalize(const float* __restrict__ stats, const float* __restrict__ gamma,
                const float* __restrict__ beta, float* __restrict__ A,
                float* __restrict__ B) {
    int c = threadIdx.x;
    const float invN = 1.0f / (float)N_NODES;
    float mean = stats[c] * invN;
    float var  = stats[EMB + c] * invN - mean * mean;
    float rstd = rsqrtf(var + EPSV);
    float a = rstd * gamma[c];
    A[c] = a;
    B[c] = beta[c] - mean * a;
}

__global__ void gcn_bn_apply(const float4* __restrict__ t, const float* __restrict__ A,
                             const float* __restrict__ B, float4* __restrict__ hout,
                             int do_relu) {
    const int n4 = N_NODES * EMB / 4;
    for (int i = blockIdx.x * blockDim.x + threadIdx.x; i < n4;
         i += gridDim.x * blockDim.x) {
        const int cb = (i & 63) * 4;
        const float4 av = *(const float4*)(A + cb);
        const float4 bv = *(const float4*)(B + cb);
        float4 v = t[i];
        v.x = v.x * av.x + bv.x;
        v.y = v.y * av.y + bv.y;
        v.z = v.z * av.z + bv.z;
        v.w = v.w * av.w + bv.w;
        if (do_relu) {
            v.x = fmaxf(v.x, 0.f); v.y = fmaxf(v.y, 0.f);
            v.z = fmaxf(v.z, 0.f); v.w = fmaxf(v.w, 0.f);
        }
        hout[i] = v;
    }
}

// ------------------------------------------------------------ pooling
__global__ void __launch_bounds__(64)
gcn_pool(const float* __restrict__ h, const int* __restrict__ batch,
         float* __restrict__ gsum) {
    const int m  = blockIdx.x;
    const int cb = threadIdx.x * 4;
    const int g  = batch[m];
    const float4 hv = *(const float4*)(h + (long)m * EMB + cb);
    float* dst = gsum + (long)g * EMB + cb;
    atomicAdd(dst + 0, hv.x);
    atomicAdd(dst + 1, hv.y);
    atomicAdd(dst + 2, hv.z);
    atomicAdd(dst + 3, hv.w);
}

__global__ void gcn_pool_cnt(const int* __restrict__ batch, float* __restrict__ cnt) {
    int m = blockIdx.x * blockDim.x + threadIdx.x;
    if (m < N_NODES) atomicAdd(&cnt[batch[m]], 1.0f);
}

__global__ void gcn_pool_div(float* __restrict__ gsum, const float* __restrict__ cnt) {
    const int n = N_GRAPHS * EMB;
    for (int i = blockIdx.x * blockDim.x + threadIdx.x; i < n;
         i += gridDim.x * blockDim.x)
        gsum[i] = gsum[i] / fmaxf(cnt[i >> 8], 1.0f);
}

// ------------------------------------------------------------ head
__global__ void __launch_bounds__(128)
gcn_head1(const float* __restrict__ g, const float* __restrict__ W,
          const float* __restrict__ b, float* __restrict__ o) {
    __shared__ float sg[EMB];
    int gid = blockIdx.x, c = threadIdx.x;
    sg[c]       = g[(long)gid * EMB + c];
    sg[c + 128] = g[(long)gid * EMB + c + 128];
    __syncthreads();
    float acc = b[c];
    #pragma unroll 8
    for (int k = 0; k < EMB; ++k) acc += sg[k] * W[c * EMB + k];
    o[(long)gid * 128 + c] = acc > 0.f ? acc : 0.f;
}

__global__ void __launch_bounds__(128)
gcn_head2(const float* __restrict__ g, const float* __restrict__ W,
          const float* __restrict__ b, float* __restrict__ o) {
    __shared__ float sg[128];
    int gid = blockIdx.x, c = threadIdx.x;
    sg[c] = g[(long)gid * 128 + c];
    __syncthreads();
    float acc = b[c];
    #pragma unroll 8
    for (int k = 0; k < 128; ++k) acc += sg[k] * W[c * 128 + k];
    o[(long)gid * 128 + c] = acc > 0.f ? acc : 0.f;
}

__global__ void __launch_bounds__(128)
gcn_head3(const float* __restrict__ g, const float* __restrict__ W,
          const float* __restrict__ b, float* __restrict__ out) {
    __shared__ float red[128];
    int gid = blockIdx.x, c = threadIdx.x;
    red[c] = g[(long)gid * 128 + c] * W[c];
    __syncthreads();
    for (int s = 64; s > 0; s >>= 1) {
        if (c < s) red[c] += red[c + s];
        __syncthreads();
    }
    if (c == 0) out[gid] = red[0] + b[0];
}

// ------------------------------------------------------------ launcher
extern "C" void kernel_launch(void* const* d_in, const int* in_sizes, int n_in,
                              void* d_out, int out_size, void* d_ws, size_t ws_size,
                              hipStream_t stream) {
    const float* x      = (const float*)d_in[0];
    const int*   ei     = (const int*)  d_in[1];
    const float* eattr  = (const float*)d_in[2];
    const int*   batch  = (const int*)  d_in[3];
    const float* xW     = (const float*)d_in[4];
    const float* xb     = (const float*)d_in[5];
    const float* eW     = (const float*)d_in[6];
    const float* eb     = (const float*)d_in[7];
    const float* gcnW   = (const float*)d_in[8];
    const float* gcnb   = (const float*)d_in[9];
    const float* rootE  = (const float*)d_in[10];
    const float* bng    = (const float*)d_in[11];
    const float* bnb    = (const float*)d_in[12];
    const float* p1W    = (const float*)d_in[13];
    const float* p1b    = (const float*)d_in[14];
    const float* p2W    = (const float*)d_in[15];
    const float* p2b    = (const float*)d_in[16];
    const float* p3W    = (const float*)d_in[17];
    const float* p3b    = (const float*)d_in[18];

    float* ws   = (float*)d_ws;
    float* h    = ws;  ws += (size_t)N_NODES * EMB;
    float* hx   = ws;  ws += (size_t)N_NODES * EMB;
    float* agg  = ws;  ws += (size_t)N_NODES * EMB;   // also holds pre-BN "t"
    float* ea   = ws;  ws += (size_t)N_EDGES * EMB;
    float* deg  = ws;  ws += N_NODES;                 // holds deg+1 after gcn_dinv
    float* dinv = ws;  ws += N_NODES;
    float* st   = ws;  ws += 2 * EMB;
    float* bnA  = ws;  ws += EMB;
    float* bnB  = ws;  ws += EMB;
    float* gsum = ws;  ws += (size_t)N_GRAPHS * EMB;
    float* gcnt = ws;  ws += N_GRAPHS;
    float* g1   = ws;  ws += (size_t)N_GRAPHS * 128;
    float* g2   = ws;  ws += (size_t)N_GRAPHS * 128;

    gcn_node_embed<<<N_NODES, EMB, 0, stream>>>(x, xW, xb, h);
    gcn_edge_embed<<<N_EDGES, EMB, 0, stream>>>(eattr, eW, eb, ea);

    gcn_zero4<<<128, 256, 0, stream>>>((float4*)deg, N_NODES / 4);
    gcn_degree<<<(N_EDGES + 255) / 256, 256, 0, stream>>>(ei, deg);
    gcn_dinv<<<(N_NODES + 255) / 256, 256, 0, stream>>>(deg, dinv);

    for (int l = 0; l < NLAYERS; ++l) {
        const float* Wl = gcnW  + (size_t)l * EMB * EMB;
        const float* bl = gcnb  + (size_t)l * EMB;
        const float* rl = rootE + (size_t)l * EMB;
        const float* gl = bng   + (size_t)l * EMB;
        const float* be = bnb   + (size_t)l * EMB;

        gcn_gemm_wmma<<<N_NODES / 32, 256, 0, stream>>>(h, Wl, bl, hx);
        gcn_zero4<<<4096, 256, 0, stream>>>((float4*)agg, N_NODES * EMB / 4);
        gcn_edge_msg<<<N_EDGES / 4, 256, 0, stream>>>(ei, dinv, hx, ea, agg);
        gcn_zero4<<<1, 128, 0, stream>>>((float4*)st, 2 * EMB / 4);
        gcn_combine_stats<<<(N_NODES + 63) / 64, 64, 0, stream>>>(h, hx, agg, rl, deg, st);
        gcn_bn_finalize<<<1, EMB, 0, stream>>>(st, gl, be, bnA, bnB);
        gcn_bn_apply<<<4096, 256, 0, stream>>>((const float4*)agg, bnA, bnB,
                                               (float4*)h, l != NLAYERS - 1);
    }

    gcn_zero4<<<1024, 256, 0, stream>>>((float4*)gsum, N_GRAPHS * EMB / 4);
    gcn_zero4<<<4, 256, 0, stream>>>((float4*)gcnt, N_GRAPHS / 4);
    gcn_pool<<<N_NODES, 64, 0, stream>>>(h, batch, gsum);
    gcn_pool_cnt<<<(N_NODES + 255) / 256, 256, 0, stream>>>(batch, gcnt);
    gcn_pool_div<<<2048, 256, 0, stream>>>(gsum, gcnt);

    gcn_head1<<<N_GRAPHS, 128, 0, stream>>>(gsum, p1W, p1b, g1);
    gcn_head2<<<N_GRAPHS, 128, 0, stream>>>(g1, p2W, p2b, g2);
    gcn_head3<<<N_GRAPHS, 128, 0, stream>>>(g2, p3W, p3b, (float*)d_out);
}